// SupervisedGraphSage_48275432407146
// MI455X (gfx1250) — compile-verified
//
#include <hip/hip_runtime.h>
#include <hip/hip_bf16.h>
#include <math.h>

typedef float v2f __attribute__((ext_vector_type(2)));
typedef float v8f __attribute__((ext_vector_type(8)));

__device__ __forceinline__ float relu_f(float x) {
    // single v_med3_f32: clamp(x, 0, +inf)
    return __builtin_amdgcn_fmed3f(x, 0.0f, __builtin_inff());
}

// ---------------------------------------------------------------------------
// Utility kernels
// ---------------------------------------------------------------------------
__global__ void zero_f32_v4(float4* __restrict__ p, long n4) {
    long i = (long)blockIdx.x * blockDim.x + threadIdx.x;
    if (i < n4) p[i] = make_float4(0.f, 0.f, 0.f, 0.f);
}

__global__ void degree_count(const int* __restrict__ dst, float* __restrict__ cnt, int E) {
    int e = blockIdx.x * blockDim.x + threadIdx.x;
    if (e < E) unsafeAtomicAdd(&cnt[dst[e]], 1.0f);
}

__global__ void invert_cnt(float* __restrict__ cnt, int n) {
    int i = blockIdx.x * blockDim.x + threadIdx.x;
    if (i < n) cnt[i] = 1.0f / fmaxf(cnt[i], 1.0f);
}

// Pad Wa1 [3,32] -> [4,32] with a zero row (enables unguarded K=4 GEMM).
__global__ void pad_w1(const float* __restrict__ W, float* __restrict__ Wp) {
    int i = blockIdx.x * blockDim.x + threadIdx.x;
    if (i < 128) Wp[i] = (i < 96) ? W[i] : 0.0f;
}

// Level-1 segment sum: one thread per edge; raw stride 3 -> padded stride 4.
__global__ void edge_scatter_add_pad(const float* __restrict__ h,
                                     const int* __restrict__ src,
                                     const int* __restrict__ dst,
                                     float* __restrict__ s, int E) {
    int e = blockIdx.x * blockDim.x + threadIdx.x;
    if (e >= E) return;
    const float* hp = h + (long)src[e] * 3;
    float*       sp = s + (long)dst[e] * 4;
#pragma unroll
    for (int f = 0; f < 3; ++f) unsafeAtomicAdd(sp + f, hp[f]);
}

// Vectorized segment sum, D = 4*DQ features per row, DQ a power of two.
// Feature chunk is the fastest-varying index: a wave's lanes sweep contiguous
// 128B runs of one edge's row (coalesced b128 gathers), then 4 f32 atomics.
template <int DQ>
__global__ void edge_scatter_add_v4(const float* __restrict__ h,
                                    const int* __restrict__ src,
                                    const int* __restrict__ dst,
                                    float* __restrict__ s, int E) {
    constexpr int D  = DQ * 4;
    constexpr int SH = (DQ == 2) ? 1 : (DQ == 4) ? 2 : (DQ == 8) ? 3 : 4;
    long idx = (long)blockIdx.x * blockDim.x + threadIdx.x;
    if (idx >= (long)E * DQ) return;
    int e = (int)(idx >> SH);
    int f = ((int)idx & (DQ - 1)) << 2;
    const float4 v = *(const float4*)(h + (long)src[e] * D + f);
    float* sp = s + (long)dst[e] * D + f;
    unsafeAtomicAdd(sp + 0, v.x);
    unsafeAtomicAdd(sp + 1, v.y);
    unsafeAtomicAdd(sp + 2, v.z);
    unsafeAtomicAdd(sp + 3, v.w);
}

// ---------------------------------------------------------------------------
// WMMA GEMM strip kernel: each wave computes a 16-row x DOUT strip of
//   O = act( diag(rowscale) * A[gather] @ W )
// via V_WMMA_F32_16X16X4_F32; W staged in LDS per block (ds_load_2addr reads).
//   A: [*, K] row-major, K % 4 == 0; rows % 16 == 0 (N = B = 50000 = 16*3125)
//   W: [K, DOUT] row-major;  O: [rows, DOUT]
// ---------------------------------------------------------------------------
template <int K, int DOUT, bool RELU, bool GATHER, bool SCALE, bool COLGUARD>
__global__ void wmma_gemm(const float* __restrict__ A,
                          const float* __restrict__ Wg,
                          float* __restrict__ O,
                          const int* __restrict__ gather,
                          const float* __restrict__ rowscale,
                          int rows) {
    constexpr int NT = (DOUT + 15) / 16;     // column tiles per strip

    __shared__ float w[K * DOUT];
    for (int i = threadIdx.x; i < K * DOUT; i += blockDim.x) w[i] = Wg[i];
    __syncthreads();

    int tm = (int)((blockIdx.x * blockDim.x + threadIdx.x) >> 5);
    if (tm * 16 >= rows) return;             // wave-uniform exit, EXEC stays full

    int lane = threadIdx.x & 31;
    int half = lane >> 4;                    // 0: lanes 0-15, 1: lanes 16-31
    int l    = lane & 15;

    // A-operand row for this lane (lanes map to matrix rows)
    int orow = tm * 16 + l;
    int arow = GATHER ? gather[orow] : orow;
    float scale = SCALE ? rowscale[arow] : 1.0f;
    const float* arow_p = A + (long)arow * K;

    v8f acc[NT];
#pragma unroll
    for (int t = 0; t < NT; ++t) acc[t] = (v8f){0.f,0.f,0.f,0.f,0.f,0.f,0.f,0.f};

#pragma unroll
    for (int k0 = 0; k0 < K; k0 += 4) {
        int ka = k0 + 2 * half;              // this lane-half's contiguous K pair
        v2f a = *(const v2f*)(arow_p + ka);  // one global_load_b64
        if (SCALE) { a.x *= scale; a.y *= scale; }
#pragma unroll
        for (int t = 0; t < NT; ++t) {
            int c = t * 16 + l;
            if (COLGUARD) c = (c < DOUT) ? c : (DOUT - 1);   // clamp: branch-free
            v2f b;
            b.x = w[ka * DOUT + c];
            b.y = w[(ka + 1) * DOUT + c];
            acc[t] = __builtin_amdgcn_wmma_f32_16x16x4_f32(
                false, a, false, b, (short)0, acc[t], false, false);
        }
    }

    // Store: acc VGPR v holds M = v + 8*half, N = lane%16
#pragma unroll
    for (int t = 0; t < NT; ++t) {
        int  c  = t * 16 + l;
        bool cv = (!COLGUARD) || (c < DOUT);
#pragma unroll
        for (int v = 0; v < 8; ++v) {
            int m = tm * 16 + v + 8 * half;
            float x = acc[t][v];
            if (RELU) x = relu_f(x);
            if (cv) O[(long)m * DOUT + c] = x;
        }
    }
}

// ---------------------------------------------------------------------------
// Host-side launchers
// ---------------------------------------------------------------------------
template <int K, int DOUT, bool RELU, bool GATHER, bool SCALE, bool COLGUARD>
static inline void launch_gemm(const float* A, const float* W, float* O,
                               const int* gather, const float* rowscale,
                               int rows, hipStream_t stream) {
    int waves  = (rows + 15) / 16;           // one wave per 16-row strip
    int blocks = (waves + 7) / 8;            // 8 waves (256 threads) per block
    wmma_gemm<K, DOUT, RELU, GATHER, SCALE, COLGUARD>
        <<<blocks, 256, 0, stream>>>(A, W, O, gather, rowscale, rows);
}

extern "C" void kernel_launch(void* const* d_in, const int* in_sizes, int n_in,
                              void* d_out, int out_size, void* d_ws, size_t ws_size,
                              hipStream_t stream) {
    const float* raw   = (const float*)d_in[0];   // [N,3]
    const int*   nodes = (const int*)  d_in[1];   // [B]
    const int*   esrc  = (const int*)  d_in[2];   // [E]
    const int*   edst  = (const int*)  d_in[3];   // [E]
    const float* Wa1 = (const float*)d_in[4];     // [3,32]
    const float* Wa2 = (const float*)d_in[5];     // [32,32]
    const float* Wa3 = (const float*)d_in[6];     // [64,64]
    const float* Wa4 = (const float*)d_in[7];     // [64,32]
    const float* We1 = (const float*)d_in[8];     // [32,32]
    const float* We2 = (const float*)d_in[9];     // [32,64]
    const float* We3 = (const float*)d_in[10];    // [64,64]
    const float* We4 = (const float*)d_in[11];    // [32,32]
    const float* Wout = (const float*)d_in[12];   // [32,40]

    const int N = in_sizes[0] / 3;
    const int B = in_sizes[1];
    const int E = in_sizes[2];

    // Workspace (floats): inv_cnt[N] | s[N*64] | abuf[N*64] | h0[N*64] | h1[N*64] | wpad[128]
    float* ws      = (float*)d_ws;
    float* inv_cnt = ws;
    float* s       = inv_cnt + N;
    float* abuf    = s    + (size_t)N * 64;
    float* h0      = abuf + (size_t)N * 64;
    float* h1      = h0   + (size_t)N * 64;
    float* wpad    = h1   + (size_t)N * 64;

    const int T = 256;
    auto blk = [](long n) { return (int)((n + 255) / 256); };

    // degree counts -> inv_cnt; pad Wa1 to 4x32
    zero_f32_v4<<<blk((N + 3) / 4), T, 0, stream>>>((float4*)inv_cnt, (N + 3) / 4);
    degree_count<<<blk(E), T, 0, stream>>>(edst, inv_cnt, E);
    invert_cnt<<<blk(N), T, 0, stream>>>(inv_cnt, N);
    pad_w1<<<1, 128, 0, stream>>>(Wa1, wpad);

    // ---- Level 1: 3 -> 32 -> 32 (K padded to 4) ----
    zero_f32_v4<<<blk((long)N), T, 0, stream>>>((float4*)s, (long)N);
    edge_scatter_add_pad<<<blk(E), T, 0, stream>>>(raw, esrc, edst, s, E);
    launch_gemm<4,  32, true, false, true,  false>(s,    wpad, abuf, nullptr, inv_cnt, N, stream);
    launch_gemm<32, 32, true, false, false, false>(abuf, We1,  h0,   nullptr, nullptr, N, stream);

    // ---- Level 2: 32 -> 32 -> 64 ----
    zero_f32_v4<<<blk((long)N * 8), T, 0, stream>>>((float4*)s, (long)N * 8);
    edge_scatter_add_v4<8><<<blk((long)E * 8), T, 0, stream>>>(h0, esrc, edst, s, E);
    launch_gemm<32, 32, true, false, true,  false>(s,    Wa2, abuf, nullptr, inv_cnt, N, stream);
    launch_gemm<32, 64, true, false, false, false>(abuf, We2, h1,   nullptr, nullptr, N, stream);

    // ---- Level 3: 64 -> 64 -> 64 ----
    zero_f32_v4<<<blk((long)N * 16), T, 0, stream>>>((float4*)s, (long)N * 16);
    edge_scatter_add_v4<16><<<blk((long)E * 16), T, 0, stream>>>(h1, esrc, edst, s, E);
    launch_gemm<64, 64, true, false, true,  false>(s,    Wa3, abuf, nullptr, inv_cnt, N, stream);
    launch_gemm<64, 64, true, false, false, false>(abuf, We3, h0,   nullptr, nullptr, N, stream);

    // ---- Level 4: 64 -> 32 -> 32 ----
    zero_f32_v4<<<blk((long)N * 16), T, 0, stream>>>((float4*)s, (long)N * 16);
    edge_scatter_add_v4<16><<<blk((long)E * 16), T, 0, stream>>>(h0, esrc, edst, s, E);
    launch_gemm<64, 32, true, false, true,  false>(s,    Wa4, abuf, nullptr, inv_cnt, N, stream);
    launch_gemm<32, 32, true, false, false, false>(abuf, We4, h1,   nullptr, nullptr, N, stream);

    // ---- Output: gather rows by `nodes`, GEMM vs W_out [32,40], no ReLU ----
    launch_gemm<32, 40, false, true, false, true>(h1, Wout, (float*)d_out,
                                                  nodes, nullptr, B, stream);
}